// CombinedSparseModel_24300924961030
// MI455X (gfx1250) — compile-verified
//
#include <hip/hip_runtime.h>
#include <hip/hip_bf16.h>

// ---------------------------------------------------------------------------
// CDNA5 / gfx1250 implementation of CombinedSparseModel.
// Dense GEMMs -> wave32 WMMA f32 16x16x4 tiles, LDS-staged K chunks
//   (templated on K/N so staging uses global_load_b128 + 32-bit addressing,
//    and all fragments are fetched before the 8-deep WMMA burst).
// Graph aggregation -> wave-per-edge atomic-add scatter (bandwidth bound).
// Attention with S=1 collapses to V/O projections (top-k + softmax = identity).
// ---------------------------------------------------------------------------

typedef __attribute__((ext_vector_type(2))) float v2f;
typedef __attribute__((ext_vector_type(8))) float v8f;

#define WAVE 32

// ---------------------------------------------------------------------------
// WMMA GEMM: C[M,N] = act(A[M,K] @ W[K,N] + bias)
// Block: 256 threads = 8 waves; block tile 32(M) x 64(N); wave tile 16x16.
// K staged through LDS in chunks of 32; 8 x v_wmma_f32_16x16x4_f32 per chunk.
// Compile-time K, N => float4 global loads, fully 32-bit index math.
// Requires M%32==0, N%64==0, K%32==0 (all shapes in this model qualify).
// ---------------------------------------------------------------------------
template<int K, int N, bool BIAS, bool RELU>
__global__ __launch_bounds__(256)
void gemm_wmma(const float* __restrict__ A, const float* __restrict__ W,
               const float* __restrict__ bias, float* __restrict__ C) {
    __shared__ float As[32][36];   // pad 4 floats: row base stays 16B aligned
    __shared__ float Bs[32][68];

    const int tid  = threadIdx.x;
    const int wave = tid >> 5;            // 0..7
    const int lane = tid & 31;
    const int rowB = blockIdx.y * 32;
    const int colB = blockIdx.x * 64;
    const int wr   = (wave >> 2) * 16;
    const int wc   = (wave & 3) * 16;

    // Staging coordinates (constant across K chunks)
    const int ar  = tid >> 3,  ac  = (tid & 7) * 4;    // A: 1 float4 / thread
    const int br0 = tid >> 4,  bc0 = (tid & 15) * 4;   // B: 2 float4 / thread
    const int br1 = (tid + 256) >> 4, bc1 = bc0;

    v8f acc = {};

    for (int k0 = 0; k0 < K; k0 += 32) {
        // Issue all global loads first (b128), then store to LDS.
        float4 av  = *(const float4*)(A + (rowB + ar) * K + k0 + ac);
        float4 bv0 = *(const float4*)(W + (k0 + br0) * N + colB + bc0);
        float4 bv1 = *(const float4*)(W + (k0 + br1) * N + colB + bc1);
        *(float4*)&As[ar][ac]   = av;
        *(float4*)&Bs[br0][bc0] = bv0;
        *(float4*)&Bs[br1][bc1] = bv1;
        __syncthreads();

        // Fragment layouts (ISA 7.12.2):
        //   A 16x4: lane -> M (mod 16); VGPR0/1 = K pair; lanes>=16 take K+2.
        //   B 4x16: lane -> N (mod 16); same K split.
        const int m  = lane & 15;
        const int kh = (lane < 16) ? 0 : 2;
        v2f af[8], bf[8];
        #pragma unroll
        for (int t = 0; t < 8; ++t) {
            const int kk = t * 4;
            af[t].x = As[wr + m][kk + kh];
            af[t].y = As[wr + m][kk + kh + 1];
            bf[t].x = Bs[kk + kh][wc + m];
            bf[t].y = Bs[kk + kh + 1][wc + m];
        }
        #pragma unroll
        for (int t = 0; t < 8; ++t)
            acc = __builtin_amdgcn_wmma_f32_16x16x4_f32(
                false, af[t], false, bf[t], (short)0, acc, false, false);
        __syncthreads();
    }

    // C/D layout: VGPR i -> M=i (lanes 0-15) or M=8+i (lanes 16-31); N = lane&15
    const int cn  = colB + wc + (lane & 15);
    const int mhi = (lane >= 16) ? 8 : 0;
    const float bcol = BIAS ? bias[cn] : 0.0f;
    #pragma unroll
    for (int i = 0; i < 8; ++i) {
        const int cm = rowB + wr + mhi + i;
        float v = acc[i] + bcol;
        if (RELU) v = fmaxf(v, 0.0f);
        C[(size_t)cm * N + cn] = v;
    }
}

// ---------------------------------------------------------------------------
// Utility / graph kernels
// ---------------------------------------------------------------------------
__global__ void fill4_kernel(float4* __restrict__ p, float v, size_t n4) {
    size_t i = (size_t)blockIdx.x * blockDim.x + threadIdx.x;
    if (i < n4) p[i] = make_float4(v, v, v, v);
}

__global__ void deg_kernel(const int* __restrict__ dst, float* __restrict__ deg, int E) {
    int e = blockIdx.x * blockDim.x + threadIdx.x;
    if (e < E) atomicAdd(&deg[dst[e]], 1.0f);
}

__global__ void rsqrt_kernel(float* __restrict__ p, int n) {
    int i = blockIdx.x * blockDim.x + threadIdx.x;
    if (i < n) p[i] = rsqrtf(p[i]);
}

// Wave per edge: out[d,:] += h[s,:] * dinv[s]*dinv[d]
template<int F>
__global__ void edge_agg_kernel(const float* __restrict__ h, const float* __restrict__ dinv,
                                const int* __restrict__ src, const int* __restrict__ dst,
                                float* __restrict__ out, int E) {
    int wave = (int)(((size_t)blockIdx.x * blockDim.x + threadIdx.x) >> 5);
    int lane = threadIdx.x & 31;
    if (wave >= E) return;
    const int s = src[wave], d = dst[wave];
    const float norm = dinv[s] * dinv[d];
    const float* hs = h + (size_t)s * F;
    float* od = out + (size_t)d * F;
    #pragma unroll
    for (int t = 0; t < F / WAVE; ++t) {
        const int f = lane + t * WAVE;
        atomicAdd(&od[f], hs[f] * norm);
    }
}

// out[n,f] += h[n,f]*dinv[n]^2 + bias[f]; optional ReLU  (self-loop + bias)
template<int LOGF, bool RELU>
__global__ void node_finalize_kernel(const float* __restrict__ h, const float* __restrict__ dinv,
                                     const float* __restrict__ bias, float* __restrict__ out,
                                     size_t total) {
    size_t i = (size_t)blockIdx.x * blockDim.x + threadIdx.x;
    if (i >= total) return;
    const size_t n = i >> LOGF;
    const int f = (int)(i & ((1u << LOGF) - 1));
    const float di = dinv[n];
    float v = out[i] + h[i] * di * di + bias[f];
    if (RELU) v = fmaxf(v, 0.0f);
    out[i] = v;
}

// Wave per node: pooled[b,:] += h2[n,:]; cnt[b] += 1
__global__ void pool_kernel(const float* __restrict__ h2, const int* __restrict__ batch,
                            float* __restrict__ pooled, float* __restrict__ cnt, int n_nodes) {
    int wave = (int)(((size_t)blockIdx.x * blockDim.x + threadIdx.x) >> 5);
    int lane = threadIdx.x & 31;
    if (wave >= n_nodes) return;
    const int b = batch[wave];
    const float* row = h2 + (size_t)wave * 128;
    float* prow = pooled + (size_t)b * 128;
    #pragma unroll
    for (int t = 0; t < 4; ++t) {
        const int f = lane + t * WAVE;
        atomicAdd(&prow[f], row[f]);
    }
    if (lane == 0) atomicAdd(&cnt[b], 1.0f);
}

__global__ void pool_div_kernel(float* __restrict__ pooled, const float* __restrict__ cnt, int B) {
    int i = blockIdx.x * blockDim.x + threadIdx.x;
    if (i >= B * 128) return;
    pooled[i] /= fmaxf(cnt[i >> 7], 1.0f);
}

__global__ void concat_kernel(const float* __restrict__ e, const float* __restrict__ g,
                              float* __restrict__ comb, int B) {
    int i = blockIdx.x * blockDim.x + threadIdx.x;
    if (i >= B * 128) return;
    const int b = i >> 7, f = i & 127;
    comb[i] = (f < 64) ? e[b * 64 + f] : g[b * 64 + (f - 64)];
}

// out[b] = dot(z[b,:64], ff_w) + ff_b
__global__ void final_dot_kernel(const float* __restrict__ z, const float* __restrict__ ff_w,
                                 const float* __restrict__ ff_b, float* __restrict__ out, int B) {
    int b = blockIdx.x * blockDim.x + threadIdx.x;
    if (b >= B) return;
    const float4* zr = (const float4*)(z + (size_t)b * 64);
    const float4* wr = (const float4*)ff_w;
    float acc = 0.0f;
    #pragma unroll
    for (int t = 0; t < 16; ++t) {
        float4 zv = zr[t], wv = wr[t];
        acc = fmaf(zv.x, wv.x, acc);
        acc = fmaf(zv.y, wv.y, acc);
        acc = fmaf(zv.z, wv.z, acc);
        acc = fmaf(zv.w, wv.w, acc);
    }
    out[b] = acc + ff_b[0];
}

// ---------------------------------------------------------------------------
// Launch orchestration
// ---------------------------------------------------------------------------
static inline dim3 grid1(size_t n, int blk) { return dim3((unsigned)((n + blk - 1) / blk)); }

extern "C" void kernel_launch(void* const* d_in, const int* in_sizes, int n_in,
                              void* d_out, int out_size, void* d_ws, size_t ws_size,
                              hipStream_t stream) {
    constexpr int B = 4096, HID = 64;
    constexpr int NN = 262144, NE = 1048576;

    const float* smiles = (const float*)d_in[0];   // [B,768]
    const float* node_x = (const float*)d_in[1];   // [NN,64]
    const int*   e_src  = (const int*)d_in[2];
    const int*   e_dst  = (const int*)d_in[3];
    const int*   batch  = (const int*)d_in[4];
    const float* emb_w1 = (const float*)d_in[5];
    const float* emb_b1 = (const float*)d_in[6];
    const float* emb_w2 = (const float*)d_in[7];
    const float* emb_b2 = (const float*)d_in[8];
    const float* g1_w   = (const float*)d_in[9];
    const float* g1_b   = (const float*)d_in[10];
    const float* g2_w   = (const float*)d_in[11];
    const float* g2_b   = (const float*)d_in[12];
    const float* gfc_w  = (const float*)d_in[13];
    const float* gfc_b  = (const float*)d_in[14];
    // q_w/q_b (15,16), k_w/k_b (17,18) are dead code at S=1
    const float* v_w    = (const float*)d_in[19];
    const float* v_b    = (const float*)d_in[20];
    const float* o_w    = (const float*)d_in[21];
    const float* o_b    = (const float*)d_in[22];
    const float* f1_w   = (const float*)d_in[23];
    const float* f1_b   = (const float*)d_in[24];
    const float* ff_w   = (const float*)d_in[25];
    const float* ff_b   = (const float*)d_in[26];
    float* out = (float*)d_out;

    // Workspace layout
    char* ws = (char*)d_ws;
    size_t off = 0;
    auto alloc = [&](size_t bytes) { float* p = (float*)(ws + off); off += (bytes + 255) & ~size_t(255); return p; };
    float* c1     = alloc((size_t)B * 1024 * 4);   // relu(smiles@w1+b1)
    float* e_emb  = alloc((size_t)B * HID * 4);    // embedding output
    float* h1     = alloc((size_t)NN * 64 * 4);    // node_x @ g1_w
    float* agg1   = alloc((size_t)NN * 64 * 4);    // conv1 output
    float* h2     = alloc((size_t)NN * 128 * 4);   // agg1 @ g2_w
    float* agg2   = alloc((size_t)NN * 128 * 4);   // conv2 output
    float* deg    = alloc((size_t)NN * 4);         // degree -> dinv (in place)
    float* pooled = alloc((size_t)B * 128 * 4);
    float* cnt    = alloc((size_t)B * 4);
    float* gvec   = alloc((size_t)B * HID * 4);
    float* comb   = alloc((size_t)B * 128 * 4);
    float* vbuf   = alloc((size_t)B * 128 * 4);
    float* abuf   = alloc((size_t)B * 128 * 4);
    float* zbuf   = alloc((size_t)B * HID * 4);

    // ---- Embedding MLP ----
    gemm_wmma<768, 1024, true, true><<<dim3(16, B / 32), 256, 0, stream>>>(smiles, emb_w1, emb_b1, c1);
    gemm_wmma<1024, 64, true, false><<<dim3(1, B / 32), 256, 0, stream>>>(c1, emb_w2, emb_b2, e_emb);

    // ---- Degree / norm (self loops give deg >= 1) ----
    fill4_kernel<<<grid1(NN / 4, 256), 256, 0, stream>>>((float4*)deg, 1.0f, NN / 4);
    deg_kernel<<<grid1(NE, 256), 256, 0, stream>>>(e_dst, deg, NE);
    rsqrt_kernel<<<grid1(NN, 256), 256, 0, stream>>>(deg, NN);   // deg -> dinv

    // ---- GCN conv 1 (64 -> 64, ReLU) ----
    gemm_wmma<64, 64, false, false><<<dim3(1, NN / 32), 256, 0, stream>>>(node_x, g1_w, nullptr, h1);
    fill4_kernel<<<grid1((size_t)NN * 16, 256), 256, 0, stream>>>((float4*)agg1, 0.0f, (size_t)NN * 16);
    edge_agg_kernel<64><<<grid1((size_t)NE * WAVE, 256), 256, 0, stream>>>(h1, deg, e_src, e_dst, agg1, NE);
    node_finalize_kernel<6, true><<<grid1((size_t)NN * 64, 256), 256, 0, stream>>>(h1, deg, g1_b, agg1, (size_t)NN * 64);

    // ---- GCN conv 2 (64 -> 128) ----
    gemm_wmma<64, 128, false, false><<<dim3(2, NN / 32), 256, 0, stream>>>(agg1, g2_w, nullptr, h2);
    fill4_kernel<<<grid1((size_t)NN * 32, 256), 256, 0, stream>>>((float4*)agg2, 0.0f, (size_t)NN * 32);
    edge_agg_kernel<128><<<grid1((size_t)NE * WAVE, 256), 256, 0, stream>>>(h2, deg, e_src, e_dst, agg2, NE);
    node_finalize_kernel<7, false><<<grid1((size_t)NN * 128, 256), 256, 0, stream>>>(h2, deg, g2_b, agg2, (size_t)NN * 128);

    // ---- Mean pool per graph ----
    fill4_kernel<<<grid1((size_t)B * 32, 256), 256, 0, stream>>>((float4*)pooled, 0.0f, (size_t)B * 32);
    fill4_kernel<<<grid1(B / 4, 256), 256, 0, stream>>>((float4*)cnt, 0.0f, B / 4);
    pool_kernel<<<grid1((size_t)NN * WAVE, 256), 256, 0, stream>>>(agg2, batch, pooled, cnt, NN);
    pool_div_kernel<<<grid1((size_t)B * 128, 256), 256, 0, stream>>>(pooled, cnt, B);

    // ---- graph FC + concat ----
    gemm_wmma<128, 64, true, false><<<dim3(1, B / 32), 256, 0, stream>>>(pooled, gfc_w, gfc_b, gvec);
    concat_kernel<<<grid1((size_t)B * 128, 256), 256, 0, stream>>>(e_emb, gvec, comb, B);

    // ---- "sparse attention" at S=1: a = (comb@Vw+Vb)@Ow+Ob ----
    gemm_wmma<128, 128, true, false><<<dim3(2, B / 32), 256, 0, stream>>>(comb, v_w, v_b, vbuf);
    gemm_wmma<128, 128, true, false><<<dim3(2, B / 32), 256, 0, stream>>>(vbuf, o_w, o_b, abuf);

    // ---- fc1 + final projection ----
    gemm_wmma<128, 64, true, false><<<dim3(1, B / 32), 256, 0, stream>>>(abuf, f1_w, f1_b, zbuf);
    final_dot_kernel<<<grid1(B, 256), 256, 0, stream>>>(zbuf, ff_w, ff_b, out, B);

    (void)in_sizes; (void)n_in; (void)out_size; (void)ws_size;
}